// ScaledDotProductAttention_72310069395864
// MI455X (gfx1250) — compile-verified
//
#include <hip/hip_runtime.h>
#include <math.h>

// ---------------------------------------------------------------------------
// Dual-softmax attention (mutual-nearest masking) for MI455X / gfx1250.
// B=4, LQ=LK=4096, D=64.  Outputs (out[B,L,D], mask[B,L,L], log_attn[B,L,L]).
// Memory-bound (541MB stores); GEMM recomputed with V_WMMA_F32_16X16X4_F32.
// ---------------------------------------------------------------------------

typedef float v2f __attribute__((ext_vector_type(2)));
typedef float v8f __attribute__((ext_vector_type(8)));

constexpr int   BB        = 4;
constexpr int   L         = 4096;
constexpr int   D         = 64;
constexpr float INV_T     = 0.125f;   // 1/8
constexpr float INV_SHARP = 40.0f;    // 1/0.025
constexpr int   ROWS      = 16;       // query rows per block
constexpr int   WAVES     = 4;        // 128 threads

// Row-wise butterfly reductions confined to a 16-lane half of a wave32.
__device__ __forceinline__ float rmax16(float v) {
  v = fmaxf(v, __shfl_xor(v, 1, 32));
  v = fmaxf(v, __shfl_xor(v, 2, 32));
  v = fmaxf(v, __shfl_xor(v, 4, 32));
  v = fmaxf(v, __shfl_xor(v, 8, 32));
  return v;
}
__device__ __forceinline__ float rsum16(float v) {
  v += __shfl_xor(v, 1, 32);
  v += __shfl_xor(v, 2, 32);
  v += __shfl_xor(v, 4, 32);
  v += __shfl_xor(v, 8, 32);
  return v;
}

// 16x16 tile of X*Y^T (unscaled) via 16 fp32 WMMAs over the D=64 depth.
// A layout (16x4 f32): lanes 0-15 supply K=4s..4s+1 of row M=lane;
// lanes 16-31 supply K=4s+2..4s+3 (hence the +2*half pointer shift).
__device__ __forceinline__ v8f score_tile(const v2f* __restrict__ a,
                                          const float* __restrict__ Yb,
                                          int kb, int n, int half) {
  const float* krow = Yb + (size_t)(kb + n) * D + 2 * half;
  v8f c = {0.f, 0.f, 0.f, 0.f, 0.f, 0.f, 0.f, 0.f};
#pragma unroll
  for (int s = 0; s < 16; ++s) {
    v2f bv = *(const v2f*)(krow + 4 * s);
    c = __builtin_amdgcn_wmma_f32_16x16x4_f32(false, a[s], false, bv,
                                              (short)0, c, false, false);
  }
  return c;
}

// ---------------------------------------------------------------------------
// Pass 1: per-row logsumexp of (X Y^T)/T.  Called with (q,k)->rowLSE and
// (k,q)->colLSE (column softmax of S == row softmax of S^T).
// ---------------------------------------------------------------------------
__global__ __launch_bounds__(128) void lse_kernel(const float* __restrict__ X,
                                                  const float* __restrict__ Y,
                                                  float* __restrict__ lse) {
  const int tid  = threadIdx.x;
  const int lane = tid & 31;
  const int wave = tid >> 5;
  const int n    = lane & 15;
  const int half = lane >> 4;
  const int blocksPerBatch = L / ROWS;
  const int b  = blockIdx.x / blocksPerBatch;
  const int q0 = (blockIdx.x % blocksPerBatch) * ROWS;

  const float* Xb = X + (size_t)b * L * D;
  const float* Yb = Y + (size_t)b * L * D;

  v2f a[16];
  {
    const float* qrow = Xb + (size_t)(q0 + n) * D + 2 * half;
#pragma unroll
    for (int s = 0; s < 16; ++s) a[s] = *(const v2f*)(qrow + 4 * s);
  }

  float rm[8], rs[8];
#pragma unroll
  for (int r = 0; r < 8; ++r) { rm[r] = -INFINITY; rs[r] = 0.0f; }

  for (int kb = wave * 16; kb < L; kb += 16 * WAVES) {
    if (kb + 16 * WAVES < L)
      __builtin_prefetch(Yb + (size_t)(kb + 16 * WAVES + n) * D, 0, 3);
    v8f c = score_tile(a, Yb, kb, n, half);
#pragma unroll
    for (int r = 0; r < 8; ++r) {
      float s_ = c[r] * INV_T;
      float tm = rmax16(s_);
      float nm = fmaxf(rm[r], tm);
      float ts = rsum16(__expf(s_ - nm));
      rs[r] = rs[r] * __expf(rm[r] - nm) + ts;
      rm[r] = nm;
    }
  }

  __shared__ float sm_m[WAVES][16];
  __shared__ float sm_s[WAVES][16];
  if (n == 0) {  // lanes 0 and 16 of each wave hold rows 0-7 / 8-15
#pragma unroll
    for (int r = 0; r < 8; ++r) {
      sm_m[wave][half * 8 + r] = rm[r];
      sm_s[wave][half * 8 + r] = rs[r];
    }
  }
  __syncthreads();
  if (tid < 16) {
    float M = -INFINITY;
#pragma unroll
    for (int w = 0; w < WAVES; ++w) M = fmaxf(M, sm_m[w][tid]);
    float S = 0.0f;
#pragma unroll
    for (int w = 0; w < WAVES; ++w) S += sm_s[w][tid] * __expf(sm_m[w][tid] - M);
    lse[(size_t)b * L + q0 + tid] = M + __logf(S);
  }
}

// ---------------------------------------------------------------------------
// Pass 2: recompute S tiles twice.
//   Sweep 1: write log_attn = S - rowLSE; online (max,sum) of
//            z = exp(2S - rowLSE - colLSE)/SHARP_T per row.
//   Sweep 2: a = exp(z - zmax)/Zsum;  mask = (a == 1/Zsum);  out += mask*V.
// ---------------------------------------------------------------------------
__global__ __launch_bounds__(128) void final_kernel(
    const float* __restrict__ Q, const float* __restrict__ Km,
    const float* __restrict__ V, const float* __restrict__ rowLSE,
    const float* __restrict__ colLSE, float* __restrict__ out,
    float* __restrict__ mask, float* __restrict__ log_attn) {
  const int tid  = threadIdx.x;
  const int lane = tid & 31;
  const int wave = tid >> 5;
  const int n    = lane & 15;
  const int half = lane >> 4;
  const int blocksPerBatch = L / ROWS;
  const int b  = blockIdx.x / blocksPerBatch;
  const int q0 = (blockIdx.x % blocksPerBatch) * ROWS;

  const float* Qb = Q  + (size_t)b * L * D;
  const float* Kb = Km + (size_t)b * L * D;
  const float* Vb = V  + (size_t)b * L * D;
  const float* rL = rowLSE + (size_t)b * L;
  const float* cL = colLSE + (size_t)b * L;

  __shared__ float out_acc[ROWS][D];
  __shared__ float sm_m[WAVES][16];
  __shared__ float sm_s[WAVES][16];
  __shared__ float ZMX[16];
  __shared__ float ZSM[16];
  for (int i = tid; i < ROWS * D; i += 128) ((float*)out_acc)[i] = 0.0f;

  v2f a[16];
  {
    const float* qrow = Qb + (size_t)(q0 + n) * D + 2 * half;
#pragma unroll
    for (int s = 0; s < 16; ++s) a[s] = *(const v2f*)(qrow + 4 * s);
  }
  float rl[8];
#pragma unroll
  for (int r = 0; r < 8; ++r) rl[r] = rL[q0 + half * 8 + r];

  // ---- sweep 1: log_attn stores + online z stats ----
  float zm[8], zs[8];
#pragma unroll
  for (int r = 0; r < 8; ++r) { zm[r] = -INFINITY; zs[r] = 0.0f; }

  for (int kb = wave * 16; kb < L; kb += 16 * WAVES) {
    if (kb + 16 * WAVES < L)
      __builtin_prefetch(Kb + (size_t)(kb + 16 * WAVES + n) * D, 0, 3);
    v8f c = score_tile(a, Kb, kb, n, half);
    float cl = cL[kb + n];
#pragma unroll
    for (int r = 0; r < 8; ++r) {
      float s_ = c[r] * INV_T;
      size_t idx = ((size_t)b * L + (q0 + half * 8 + r)) * L + kb + n;
      log_attn[idx] = s_ - rl[r];
      float z  = __expf(2.0f * s_ - rl[r] - cl) * INV_SHARP;
      float tm = rmax16(z);
      float nm = fmaxf(zm[r], tm);
      float ts = rsum16(__expf(z - nm));
      zs[r] = zs[r] * __expf(zm[r] - nm) + ts;
      zm[r] = nm;
    }
  }

  if (n == 0) {
#pragma unroll
    for (int r = 0; r < 8; ++r) {
      sm_m[wave][half * 8 + r] = zm[r];
      sm_s[wave][half * 8 + r] = zs[r];
    }
  }
  __syncthreads();
  if (tid < 16) {
    float M = -INFINITY;
#pragma unroll
    for (int w = 0; w < WAVES; ++w) M = fmaxf(M, sm_m[w][tid]);
    float S = 0.0f;
#pragma unroll
    for (int w = 0; w < WAVES; ++w) S += sm_s[w][tid] * __expf(sm_m[w][tid] - M);
    ZMX[tid] = M;
    ZSM[tid] = S;
  }
  __syncthreads();

  float Zm[8], Zs[8], Am[8];
#pragma unroll
  for (int r = 0; r < 8; ++r) {
    Zm[r] = ZMX[half * 8 + r];
    Zs[r] = ZSM[half * 8 + r];
    Am[r] = 1.0f / Zs[r];  // a_max: numerator max is exactly exp(0)=1
  }

  // ---- sweep 2: mask stores + out accumulation ----
  for (int kb = wave * 16; kb < L; kb += 16 * WAVES) {
    if (kb + 16 * WAVES < L)
      __builtin_prefetch(Kb + (size_t)(kb + 16 * WAVES + n) * D, 0, 3);
    v8f c = score_tile(a, Kb, kb, n, half);
    float cl = cL[kb + n];
#pragma unroll
    for (int r = 0; r < 8; ++r) {
      float s_ = c[r] * INV_T;
      float z  = __expf(2.0f * s_ - rl[r] - cl) * INV_SHARP;
      float aa = __expf(z - Zm[r]) / Zs[r];
      float mk = (aa == Am[r]) ? 1.0f : 0.0f;
      size_t idx = ((size_t)b * L + (q0 + half * 8 + r)) * L + kb + n;
      mask[idx] = mk;
      if (mk != 0.0f) {  // ~1 hit per row: LDS float atomics, nearly free
        const float* vrow = Vb + (size_t)(kb + n) * D;
#pragma unroll 4
        for (int d = 0; d < D; ++d)
          atomicAdd(&out_acc[half * 8 + r][d], vrow[d]);
      }
    }
  }
  __syncthreads();

  for (int i = tid; i < ROWS * D; i += 128) {
    int m = i >> 6, d = i & 63;
    out[((size_t)b * L + q0 + m) * D + d] = out_acc[m][d];
  }
}

// ---------------------------------------------------------------------------
extern "C" void kernel_launch(void* const* d_in, const int* in_sizes, int n_in,
                              void* d_out, int out_size, void* d_ws,
                              size_t ws_size, hipStream_t stream) {
  const float* q = (const float*)d_in[0];
  const float* k = (const float*)d_in[1];
  const float* v = (const float*)d_in[2];

  float* out      = (float*)d_out;                 // B*L*D
  float* mask     = out + (size_t)BB * L * D;      // B*L*L
  float* log_attn = mask + (size_t)BB * L * L;     // B*L*L

  float* rowLSE = (float*)d_ws;                    // BB*L floats
  float* colLSE = rowLSE + (size_t)BB * L;         // BB*L floats (128KB total)

  dim3 grid(BB * (L / ROWS));
  dim3 block(WAVES * 32);

  lse_kernel<<<grid, block, 0, stream>>>(q, k, rowLSE);   // row stats of S
  lse_kernel<<<grid, block, 0, stream>>>(k, q, colLSE);   // col stats (S^T)
  final_kernel<<<grid, block, 0, stream>>>(q, k, v, rowLSE, colLSE, out, mask,
                                           log_attn);
}